// Net1_69114613728512
// MI455X (gfx1250) — compile-verified
//
#include <hip/hip_runtime.h>

typedef __attribute__((ext_vector_type(2))) float v2f;
typedef __attribute__((ext_vector_type(4))) float v4f;
typedef __attribute__((ext_vector_type(8))) float v8f;

#define M_TOTAL (16 * 4096)   // B*N rows
#define K_COLS  512           // output columns (= rows of C)
#define D_DIM   64            // reduction dim
#define ROWS_PER_WG   128     // 8 waves * 16 rows
#define CHUNK_ROWS    128     // C rows staged in LDS per chunk
#define NTILES_CHUNK  (CHUNK_ROWS / 16)   // 8
#define NCHUNKS       (K_COLS / CHUNK_ROWS) // 4

__global__ __launch_bounds__(256)
void sl2_wmma_kernel(const float* __restrict__ x,
                     const float* __restrict__ C,
                     const float* __restrict__ S,
                     float* __restrict__ out) {
    // C chunk staged in WMMA-B fragment layout: frag[(t*16+s)*32 + lane] is the
    // float2 {C[n][4s+2h], C[n][4s+2h+1]} with n = chunkbase + t*16 + (lane&15),
    // h = lane>>4.  One conflict-free ds_load_b64 per lane per WMMA step.
    __shared__ v2f   frag[CHUNK_ROWS * 32];   // 32 KB
    __shared__ float c2s[K_COLS];             // 2 KB  (||C_k||^2)
    __shared__ float ss[K_COLS];              // 2 KB  (S[k])

    const int tid   = threadIdx.x;
    const int wave  = tid >> 5;
    const int lane  = tid & 31;
    const int lan15 = lane & 15;
    const int h     = lane >> 4;      // lane half selects K pair {0,1} vs {2,3}

    // ---- stage ||C_k||^2 and S[k] (2 rows per thread) ----
    #pragma unroll
    for (int r = 0; r < 2; ++r) {
        int k = tid * 2 + r;
        const float* cp = C + k * D_DIM;
        float a2 = 0.f;
        #pragma unroll
        for (int d = 0; d < D_DIM; d += 4) {
            v4f c4 = *(const v4f*)(cp + d);
            a2 = fmaf(c4.x, c4.x, a2);
            a2 = fmaf(c4.y, c4.y, a2);
            a2 = fmaf(c4.z, c4.z, a2);
            a2 = fmaf(c4.w, c4.w, a2);
        }
        c2s[k] = a2;
        ss[k]  = S[k];
    }

    // ---- load A fragments (this wave's 16 x-rows) into registers ----
    // 32-bit A 16x4 layout: lane holds row (lane&15); VGPR0=K(2h), VGPR1=K(2h+1)
    const int rowbase = blockIdx.x * ROWS_PER_WG + wave * 16;
    const float* xrow = x + (rowbase + lan15) * D_DIM + 2 * h;
    v2f a[16];
    float part = 0.f;
    #pragma unroll
    for (int s = 0; s < 16; ++s) {
        a[s] = *(const v2f*)(xrow + 4 * s);
        part = fmaf(a[s].x, a[s].x, part);
        part = fmaf(a[s].y, a[s].y, part);
    }
    // lane l and l^16 hold complementary halves of row (l&15)'s d-range
    float x2full = part + __shfl_xor(part, 16, 32);   // x2[row lane&15]
    // C/D layout: VGPR j holds M = j + 8*h  -> broadcast the needed x2 per j
    float xb[8];
    #pragma unroll
    for (int j = 0; j < 8; ++j)
        xb[j] = __shfl(x2full, j + 8 * h, 32);

    // ---- loop over C chunks ----
    for (int cc = 0; cc < NCHUNKS; ++cc) {
        __syncthreads();   // frag safe to overwrite
        // coalesced float2 global reads, scatter into fragment layout
        const v2f* csrc = (const v2f*)(C + cc * CHUNK_ROWS * D_DIM);
        #pragma unroll
        for (int it = 0; it < (CHUNK_ROWS * 32) / 256; ++it) {
            int q = it * 256 + tid;            // float2 index within chunk
            v2f val = csrc[q];
            int nloc = q >> 5;                 // local C row 0..127
            int d    = (q & 31) * 2;           // even d 0..62
            int s    = d >> 2;
            int hh   = (d >> 1) & 1;
            int f    = (((nloc >> 4) * 16 + s) << 5) + (nloc & 15) + (hh << 4);
            frag[f] = val;
        }
        __syncthreads();

        for (int t = 0; t < NTILES_CHUNK; ++t) {
            int   kcol = cc * CHUNK_ROWS + t * 16 + lan15;  // output column (N=lane&15)
            float sn   = ss[kcol];
            float c2n  = c2s[kcol];

            v8f acc0 = {};
            v8f acc1 = {};
            const v2f* fb = &frag[t * 16 * 32];
            #pragma unroll
            for (int s = 0; s < 16; s += 2) {
                v2f b0 = fb[(s + 0) * 32 + lane];
                v2f b1 = fb[(s + 1) * 32 + lane];
                acc0 = __builtin_amdgcn_wmma_f32_16x16x4_f32(
                           false, a[s + 0], false, b0, (short)0, acc0, false, false);
                acc1 = __builtin_amdgcn_wmma_f32_16x16x4_f32(
                           false, a[s + 1], false, b1, (short)0, acc1, false, false);
            }

            // SL = S*(x2 + c2) - 2*S*xc
            float m2s  = -2.0f * sn;
            float base = sn * c2n;
            float* op = out + (rowbase + 8 * h) * K_COLS + kcol;
            #pragma unroll
            for (int j = 0; j < 8; ++j) {
                float xc = acc0[j] + acc1[j];
                op[j * K_COLS] = fmaf(m2s, xc, fmaf(sn, xb[j], base));
            }
        }
    }
}

extern "C" void kernel_launch(void* const* d_in, const int* in_sizes, int n_in,
                              void* d_out, int out_size, void* d_ws, size_t ws_size,
                              hipStream_t stream) {
    (void)in_sizes; (void)n_in; (void)out_size; (void)d_ws; (void)ws_size;
    const float* x = (const float*)d_in[0];
    const float* C = (const float*)d_in[1];
    const float* S = (const float*)d_in[2];
    float* out = (float*)d_out;

    dim3 grid(M_TOTAL / ROWS_PER_WG);   // 512 blocks
    dim3 block(256);                    // 8 wave32
    hipLaunchKernelGGL(sl2_wmma_kernel, grid, block, 0, stream, x, C, S, out);
}